// MyNet_23510650978645
// MI455X (gfx1250) — compile-verified
//
#include <hip/hip_runtime.h>
#include <math.h>

typedef __attribute__((ext_vector_type(2))) float v2f;
typedef __attribute__((ext_vector_type(8))) float v8f;

#define HW1024 (1024*1024)
#define HW512  (512*512)
#define HW256  (256*256)

// ---------------------------------------------------------------------------
// Fused: enc conv(3x3,3->24,pad1) + embed conv(1x1,24->32, f32 WMMA) + maxpool2
// One wave = 16 output pixels (one row segment at 512 res) x 32 couts.
// WMMA mapping (V_WMMA_F32_16X16X4_F32, D = A(16x4) x B(4x16) + C):
//   A = embed weights  [cout=M][cin=K], lane: M=lane%16, K = v + 2*(lane>=16)
//   B = enc activations[cin=K][pix=N], lane: N=lane%16, K = v + 2*(lane>=16)
//   D: vgpr r, lane l -> cout = mt*16 + r + 8*(l>=16), pix = l%16  (coalesced)
// ---------------------------------------------------------------------------
__global__ __launch_bounds__(256) void k_coding(
    const float* __restrict__ x1, const float* __restrict__ x2,
    const float* __restrict__ enc_w, const float* __restrict__ enc_b,
    const float* __restrict__ emb_w, const float* __restrict__ emb_b,
    float* __restrict__ c1, float* __restrict__ c2)
{
    __shared__ float sw[24*27];
    __shared__ float sb[24];
    __shared__ float sew[32*24];
    __shared__ float seb[32];
    int tid = threadIdx.x;
    for (int i = tid; i < 24*27; i += 256) sw[i] = enc_w[i];
    for (int i = tid; i < 32*24; i += 256) sew[i] = emb_w[i];
    if (tid < 24) sb[tid] = enc_b[tid];
    if (tid < 32) seb[tid] = emb_b[tid];
    __syncthreads();

    const float* xin = (blockIdx.y == 0) ? x1 : x2;
    float* cbuf      = (blockIdx.y == 0) ? c1 : c2;

    int wave = tid >> 5;
    int lane = tid & 31;
    int lm   = lane & 15;
    int h2   = (lane >> 4) << 1;          // 0 or 2 : K offset for upper half-wave

    int tile = blockIdx.x * 8 + wave;     // 16384 tiles per image
    int cy   = tile >> 5;                 // 512/16 = 32 tiles per row
    int cx0  = (tile & 31) << 4;
    int col  = cx0 + lm;                  // output column 0..511

    // 3ch x 4x4 input patch covering both pool positions + 3x3 halo
    int iy0 = 2*cy - 1;
    int ix0 = 2*col - 1;
    float patch[3][4][4];
    #pragma unroll
    for (int ic = 0; ic < 3; ++ic)
      #pragma unroll
      for (int ry = 0; ry < 4; ++ry)
        #pragma unroll
        for (int rx = 0; rx < 4; ++rx) {
            int y = iy0 + ry, x = ix0 + rx;
            patch[ic][ry][rx] = (y >= 0 && y < 1024 && x >= 0 && x < 1024)
                              ? xin[ic*HW1024 + y*1024 + x] : 0.0f;
        }

    // enc activations in WMMA-B register layout: encv[pos][kstep] = {ch 4t+h2, ch 4t+1+h2}
    v2f encv[4][6];
    #pragma unroll
    for (int t = 0; t < 6; ++t) {
        #pragma unroll
        for (int v = 0; v < 2; ++v) {
            int ch = 4*t + v + h2;
            float w27[27];
            #pragma unroll
            for (int q = 0; q < 27; ++q) w27[q] = sw[ch*27 + q];
            float bias = sb[ch];
            #pragma unroll
            for (int p = 0; p < 4; ++p) {
                int dy = p >> 1, dx = p & 1;
                float acc = bias;
                #pragma unroll
                for (int ic = 0; ic < 3; ++ic)
                  #pragma unroll
                  for (int ky = 0; ky < 3; ++ky)
                    #pragma unroll
                    for (int kx = 0; kx < 3; ++kx)
                        acc += patch[ic][dy+ky][dx+kx] * w27[ic*9 + ky*3 + kx];
                encv[p][t][v] = acc;
            }
        }
    }

    // embed GEMM + elementwise max over the 4 pooled positions
    #pragma unroll
    for (int mt = 0; mt < 2; ++mt) {
        int mrow = mt*16 + lm;
        v2f a[6];
        #pragma unroll
        for (int t = 0; t < 6; ++t) {
            a[t][0] = sew[mrow*24 + 4*t + h2 + 0];
            a[t][1] = sew[mrow*24 + 4*t + h2 + 1];
        }
        v8f macc;
        #pragma unroll
        for (int p = 0; p < 4; ++p) {
            v8f acc = {0.f,0.f,0.f,0.f,0.f,0.f,0.f,0.f};
            #pragma unroll
            for (int t = 0; t < 6; ++t)
                acc = __builtin_amdgcn_wmma_f32_16x16x4_f32(
                        false, a[t], false, encv[p][t], (short)0, acc, false, false);
            if (p == 0) macc = acc;
            else {
                #pragma unroll
                for (int r = 0; r < 8; ++r) macc[r] = fmaxf(macc[r], acc[r]);
            }
        }
        #pragma unroll
        for (int r = 0; r < 8; ++r) {
            int co = mt*16 + r + (h2 << 2);       // +8 for upper half-wave
            cbuf[co*HW512 + cy*512 + col] = macc[r] + seb[co];
        }
    }
}

// ---------------------------------------------------------------------------
// pixel_match: 20 channels out: [0..8] pm(a,b), [9..17] pm(b,a), [18..19] pos
// ---------------------------------------------------------------------------
__global__ __launch_bounds__(256) void k_pm(
    const float* __restrict__ a, const float* __restrict__ b,
    float* __restrict__ out, int H, int W)
{
    int idx = blockIdx.x * 256 + threadIdx.x;
    int HWl = H * W;
    if (idx >= HWl) return;
    int h = idx / W, w = idx % W;
    float dab[9], dba[9];
    if (h >= 1 && h <= H-2 && w >= 1 && w <= W-2) {
        float ac[32], bc[32];
        #pragma unroll
        for (int c = 0; c < 32; ++c) { ac[c] = a[c*HWl + idx]; bc[c] = b[c*HWl + idx]; }
        float suma = 0.f, sumb = 0.f;
        #pragma unroll
        for (int d = 0; d < 9; ++d) {
            int off = idx + (d/3 - 1)*W + (d%3 - 1);
            float sab = 0.f, sba = 0.f;
            #pragma unroll
            for (int c = 0; c < 32; ++c) {
                float av = a[c*HWl + off];
                float bv = b[c*HWl + off];
                float t1 = av - bc[c];
                float t2 = bv - ac[c];
                sab += t1*t1;
                sba += t2*t2;
            }
            dab[d] = 1.0f / sqrtf(sab + 0.01f);
            dba[d] = 1.0f / sqrtf(sba + 0.01f);
            suma += dab[d]; sumb += dba[d];
        }
        #pragma unroll
        for (int d = 0; d < 9; ++d) { dab[d] /= suma; dba[d] /= sumb; }
    } else {
        #pragma unroll
        for (int d = 0; d < 9; ++d) { dab[d] = 1.0f/9.0f; dba[d] = 1.0f/9.0f; }
    }
    #pragma unroll
    for (int d = 0; d < 9; ++d) {
        out[d*HWl + idx]     = dab[d];
        out[(9+d)*HWl + idx] = dba[d];
    }
    out[18*HWl + idx] = 2.0f * (float)h / (float)(H-1) - 1.0f;
    out[19*HWl + idx] = 2.0f * (float)w / (float)(W-1) - 1.0f;
}

// ---------------------------------------------------------------------------
__global__ __launch_bounds__(256) void k_maxpool(
    const float* __restrict__ in, float* __restrict__ out, int C, int OH, int OW)
{
    int idx = blockIdx.x * 256 + threadIdx.x;
    if (idx >= C*OH*OW) return;
    int c = idx / (OH*OW), rem = idx % (OH*OW);
    int oy = rem / OW, ox = rem % OW;
    int IW = 2*OW;
    const float* p = in + c*(4*OH*OW) + (2*oy)*IW + 2*ox;
    out[idx] = fmaxf(fmaxf(p[0], p[1]), fmaxf(p[IW], p[IW+1]));
}

__global__ __launch_bounds__(256) void k_pool_pair(
    const float* __restrict__ c1, const float* __restrict__ c2,
    float* __restrict__ p1, float* __restrict__ p2, float* __restrict__ pd)
{
    int idx = blockIdx.x * 256 + threadIdx.x;       // 32*256*256 exact
    int c = idx >> 16, rem = idx & 0xFFFF;
    int oy = rem >> 8, ox = rem & 0xFF;
    const float* a = c1 + c*HW512 + (2*oy)*512 + 2*ox;
    const float* b = c2 + c*HW512 + (2*oy)*512 + 2*ox;
    float v1 = fmaxf(fmaxf(a[0], a[1]), fmaxf(a[512], a[513]));
    float v2 = fmaxf(fmaxf(b[0], b[1]), fmaxf(b[512], b[513]));
    p1[idx] = v1; p2[idx] = v2; pd[idx] = v1 - v2;
}

// ---------------------------------------------------------------------------
// dense conv 5x5, 32->8, pad 2, leaky_relu(0.01), on pdiff @256
// ---------------------------------------------------------------------------
__global__ __launch_bounds__(256) void k_dense(
    const float* __restrict__ pd, const float* __restrict__ w,
    const float* __restrict__ bias, float* __restrict__ out)
{
    __shared__ float swt[8*32*25];
    __shared__ float sbi[8];
    int tid = threadIdx.x;
    for (int i = tid; i < 8*32*25; i += 256) swt[i] = w[i];
    if (tid < 8) sbi[tid] = bias[tid];
    __syncthreads();

    int idx = blockIdx.x * 256 + tid;               // 65536 exact
    int y = idx >> 8, x = idx & 0xFF;
    float acc[8];
    #pragma unroll
    for (int oc = 0; oc < 8; ++oc) acc[oc] = sbi[oc];
    for (int ky = 0; ky < 5; ++ky) {
        int iy = y - 2 + ky;
        if (iy < 0 || iy >= 256) continue;
        for (int kx = 0; kx < 5; ++kx) {
            int ix = x - 2 + kx;
            if (ix < 0 || ix >= 256) continue;
            int kidx = ky*5 + kx;
            #pragma unroll
            for (int ic = 0; ic < 32; ++ic) {
                float v = pd[ic*HW256 + iy*256 + ix];
                #pragma unroll
                for (int oc = 0; oc < 8; ++oc)
                    acc[oc] += v * swt[(oc*32 + ic)*25 + kidx];
            }
        }
    }
    #pragma unroll
    for (int oc = 0; oc < 8; ++oc) {
        float v = acc[oc];
        out[oc*HW256 + idx] = (v > 0.0f) ? v : 0.01f*v;
    }
}

// ---------------------------------------------------------------------------
// f1 conv 1x1, 48->64 via WMMA f32 16x16x4. Wave = 16 pixels x 64 couts.
// ---------------------------------------------------------------------------
__global__ __launch_bounds__(256) void k_f1(
    const float* __restrict__ xc, const float* __restrict__ w,
    const float* __restrict__ bias, float* __restrict__ y)
{
    __shared__ float swt[64*48];
    __shared__ float sbi[64];
    int tid = threadIdx.x;
    for (int i = tid; i < 64*48; i += 256) swt[i] = w[i];
    if (tid < 64) sbi[tid] = bias[tid];
    __syncthreads();

    int lane = tid & 31;
    int lm   = lane & 15;
    int h2   = (lane >> 4) << 1;
    int px0  = (blockIdx.x * 8 + (tid >> 5)) * 16;   // 4096 waves exact

    v2f bm[12];
    #pragma unroll
    for (int t = 0; t < 12; ++t) {
        bm[t][0] = xc[(4*t + h2 + 0)*HW256 + px0 + lm];
        bm[t][1] = xc[(4*t + h2 + 1)*HW256 + px0 + lm];
    }
    #pragma unroll
    for (int mt = 0; mt < 4; ++mt) {
        int mrow = mt*16 + lm;
        v2f a[12];
        #pragma unroll
        for (int t = 0; t < 12; ++t) {
            a[t][0] = swt[mrow*48 + 4*t + h2 + 0];
            a[t][1] = swt[mrow*48 + 4*t + h2 + 1];
        }
        v8f acc = {0.f,0.f,0.f,0.f,0.f,0.f,0.f,0.f};
        #pragma unroll
        for (int t = 0; t < 12; ++t)
            acc = __builtin_amdgcn_wmma_f32_16x16x4_f32(
                    false, a[t], false, bm[t], (short)0, acc, false, false);
        #pragma unroll
        for (int r = 0; r < 8; ++r) {
            int co = mt*16 + r + (h2 << 2);
            y[co*HW256 + px0 + lm] = acc[r] + sbi[co];
        }
    }
}

// ---------------------------------------------------------------------------
__global__ __launch_bounds__(256) void k_amp3(const float* __restrict__ y,
                                              float* __restrict__ amp)
{
    int c = blockIdx.x / 9, r = blockIdx.x % 9;
    int i = r / 3, j = r % 3;
    int hs = (i*256)/3,   he = ((i+1)*256 + 2)/3;
    int ws = (j*256)/3,   we = ((j+1)*256 + 2)/3;
    int nh = he - hs, nw = we - ws, n = nh*nw;
    const float* p = y + c*HW256;
    float m = -INFINITY;
    for (int t = threadIdx.x; t < n; t += 256)
        m = fmaxf(m, p[(hs + t/nw)*256 + (ws + t%nw)]);
    __shared__ float red[256];
    red[threadIdx.x] = m;
    __syncthreads();
    for (int s = 128; s > 0; s >>= 1) {
        if (threadIdx.x < s) red[threadIdx.x] = fmaxf(red[threadIdx.x], red[threadIdx.x+s]);
        __syncthreads();
    }
    if (threadIdx.x == 0) amp[c*9 + r] = red[0];
}

// ---------------------------------------------------------------------------
// head: f2 conv(3x3 valid on 3x3 -> 128), l2-ish norm, f3 linear, 3x3 assembly
// ---------------------------------------------------------------------------
__global__ __launch_bounds__(128) void k_head(
    const float* __restrict__ amp, const float* __restrict__ f2w,
    const float* __restrict__ f2b, const float* __restrict__ f3w,
    const float* __restrict__ f3b, float* __restrict__ out)
{
    __shared__ float sx[128];
    __shared__ float red[128];
    __shared__ float v6[6];
    __shared__ float sscale;
    int t = threadIdx.x;
    float acc = f2b[t];
    for (int c = 0; c < 64; ++c) {
        #pragma unroll
        for (int k = 0; k < 9; ++k)
            acc += amp[c*9 + k] * f2w[(t*64 + c)*9 + k];
    }
    sx[t] = acc;
    red[t] = acc * acc;
    __syncthreads();
    for (int s = 64; s > 0; s >>= 1) {
        if (t < s) red[t] += red[t + s];
        __syncthreads();
    }
    if (t == 0) sscale = 1.0f / sqrtf(red[0] + 0.001f);
    __syncthreads();
    if (t < 6) {
        float v = f3b[t];
        for (int k = 0; k < 128; ++k) v += (sx[k] * sscale) * f3w[t*128 + k];
        v6[t] = v;
    }
    __syncthreads();
    if (t < 9) {
        int row = t / 3, col = t % 3;
        float raw = (row < 2) ? v6[row*3 + col] : 0.0f;
        out[t] = ((row == col) ? 1.0f : 0.0f)
               + fminf(fmaxf(raw, -0.11f), 0.11f) + 0.01f * raw;
    }
}

// ---------------------------------------------------------------------------
extern "C" void kernel_launch(void* const* d_in, const int* in_sizes, int n_in,
                              void* d_out, int out_size, void* d_ws, size_t ws_size,
                              hipStream_t stream) {
    (void)in_sizes; (void)n_in; (void)out_size; (void)ws_size;
    const float* x1    = (const float*)d_in[0];
    const float* x2    = (const float*)d_in[1];
    const float* enc_w = (const float*)d_in[2];
    const float* enc_b = (const float*)d_in[3];
    const float* emb_w = (const float*)d_in[4];
    const float* emb_b = (const float*)d_in[5];
    const float* dns_w = (const float*)d_in[6];
    const float* dns_b = (const float*)d_in[7];
    const float* f1_w  = (const float*)d_in[8];
    const float* f1_b  = (const float*)d_in[9];
    const float* f2_w  = (const float*)d_in[10];
    const float* f2_b  = (const float*)d_in[11];
    const float* f3_w  = (const float*)d_in[12];
    const float* f3_b  = (const float*)d_in[13];

    float* ws    = (float*)d_ws;
    float* c1    = ws;                       // 32*512*512   = 8388608
    float* c2    = ws + 8388608;             // 32*512*512
    float* pm512 = ws + 16777216;            // 20*512*512   = 5242880
    float* p1    = ws + 22020096;            // 32*256*256   = 2097152
    float* p2    = ws + 24117248;
    float* pdif  = ws + 26214400;
    float* xcat  = ws + 28311552;            // 48*256*256   = 3145728
    float* amp   = ws + 31457280;            // 576
    float* y1    = ws;                       // 64*256*256, reuses dead c1 region

    k_coding  <<<dim3(2048, 2), 256, 0, stream>>>(x1, x2, enc_w, enc_b, emb_w, emb_b, c1, c2);
    k_pm      <<<HW512/256, 256, 0, stream>>>(c1, c2, pm512, 512, 512);
    k_maxpool <<<(20*HW256)/256, 256, 0, stream>>>(pm512, xcat, 20, 256, 256);        // level1 -> ch 0..19
    k_pool_pair<<<(32*HW256)/256, 256, 0, stream>>>(c1, c2, p1, p2, pdif);
    k_pm      <<<HW256/256, 256, 0, stream>>>(p1, p2, xcat + 20*HW256, 256, 256);     // level2 -> ch 20..39
    k_dense   <<<HW256/256, 256, 0, stream>>>(pdif, dns_w, dns_b, xcat + 40*HW256);   // dense  -> ch 40..47
    k_f1      <<<512, 256, 0, stream>>>(xcat, f1_w, f1_b, y1);
    k_amp3    <<<576, 256, 0, stream>>>(y1, amp);
    k_head    <<<1, 128, 0, stream>>>(amp, f2_w, f2_b, f3_w, f3_b, (float*)d_out);
}